// SelectiveStateSpaceModel_3779571220940
// MI455X (gfx1250) — compile-verified
//
#include <hip/hip_runtime.h>
#include <hip/hip_bf16.h>
#include <math.h>

// ---------------------------------------------------------------------------
// Mamba-style selective SSM on MI455X (gfx1250, wave32).
//   L = 4096, D_INNER = 4096, DT_RANK = 128, D_STATE = 16
// Phase 1: dbc = x @ W_in^T          (fp32 WMMA 16x16x4)
// Phase 2: delta = softplus(dbc_d @ W_delta^T)  (fp32 WMMA, fused softplus)
// Phase 3: sequential-in-L scan, parallel over (channel, state)
// ---------------------------------------------------------------------------

#define SEQ_LEN   4096
#define D_INNER   4096
#define DT_RANK   128
#define D_STATE   16
#define NCOLS     (DT_RANK + 2 * D_STATE)   // 160

typedef __attribute__((ext_vector_type(2))) float v2f;
typedef __attribute__((ext_vector_type(8))) float v8f;

// ---------------------------------------------------------------------------
// Kernel 1: dbc = x @ W_in^T, routed into delta_raw [L,128], Bm [L,16], Cm [L,16]
// One wave per 16x16 output tile. K = 4096 in steps of 4 (V_WMMA_F32_16X16X4_F32).
//
// fp32 A-matrix 16x4 layout (ISA 7.12.2): lane l<16 holds {A[l][0],A[l][1]},
// lane 16+l holds {A[l][2],A[l][3]}.  B 4x16: lanes 0-15 hold rows K=0,1;
// lanes 16-31 hold rows K=2,3 (row striped across lanes, one row pair / VGPR).
// C/D: VGPR v, lane l -> (M = v + 8*(l>=16), N = l&15).
// ---------------------------------------------------------------------------
__global__ __launch_bounds__(32) void ssm_gemm_in(
    const float* __restrict__ x,      // [L, D_INNER] row-major
    const float* __restrict__ W_in,   // [160, D_INNER] row-major
    float* __restrict__ dRaw,         // [L, DT_RANK]
    float* __restrict__ Bm,           // [L, D_STATE]
    float* __restrict__ Cm)           // [L, D_STATE]
{
    const int lane = threadIdx.x;          // 0..31
    const int m    = lane & 15;            // row/col within tile
    const int kk   = lane >> 4;            // K-subgroup (0/1)
    const int tM   = blockIdx.x;           // 0..255  (L/16)
    const int tN   = blockIdx.y;           // 0..9    (160/16)

    const size_t arow = (size_t)(tM * 16 + m) * D_INNER;  // A row (x)
    const size_t brow = (size_t)(tN * 16 + m) * D_INNER;  // B row (W_in row = out col)

    v8f acc = {};
#pragma unroll 4
    for (int k = 0; k < D_INNER; k += 4) {
        v2f a = *(const v2f*)(x    + arow + k + 2 * kk);  // A[m][k+2kk .. +1]
        v2f b = *(const v2f*)(W_in + brow + k + 2 * kk);  // B[k+2kk..][n=m] == W_in[col][k..]
        acc = __builtin_amdgcn_wmma_f32_16x16x4_f32(
            /*neg_a=*/false, a, /*neg_b=*/false, b,
            /*c_mod=*/(short)0, acc, /*reuse_a=*/false, /*reuse_b=*/false);
    }

    const int c = tN * 16 + m;   // global output column 0..159
#pragma unroll
    for (int v = 0; v < 8; ++v) {
        const int r = tM * 16 + v + 8 * kk;               // global output row
        const float val = acc[v];
        if (tN < 8) {
            dRaw[(size_t)r * DT_RANK + c] = val;
        } else if (tN == 8) {
            Bm[(size_t)r * D_STATE + (c - DT_RANK)] = val;
        } else {
            Cm[(size_t)r * D_STATE + (c - DT_RANK - D_STATE)] = val;
        }
    }
}

// ---------------------------------------------------------------------------
// Kernel 2: delta = softplus(dRaw @ W_delta^T)   [L, D_INNER], K = 128
// ---------------------------------------------------------------------------
__global__ __launch_bounds__(32) void ssm_gemm_delta(
    const float* __restrict__ dRaw,     // [L, 128]
    const float* __restrict__ W_delta,  // [D_INNER, 128]
    float* __restrict__ delta)          // [L, D_INNER]
{
    const int lane = threadIdx.x;
    const int m    = lane & 15;
    const int kk   = lane >> 4;
    const int tM   = blockIdx.x;        // 0..255
    const int tN   = blockIdx.y;        // 0..255

    const size_t arow = (size_t)(tM * 16 + m) * DT_RANK;
    const size_t brow = (size_t)(tN * 16 + m) * DT_RANK;

    v8f acc = {};
#pragma unroll
    for (int k = 0; k < DT_RANK; k += 4) {
        v2f a = *(const v2f*)(dRaw    + arow + k + 2 * kk);
        v2f b = *(const v2f*)(W_delta + brow + k + 2 * kk);
        acc = __builtin_amdgcn_wmma_f32_16x16x4_f32(
            false, a, false, b, (short)0, acc, false, false);
    }

    const int c = tN * 16 + m;
#pragma unroll
    for (int v = 0; v < 8; ++v) {
        const int r = tM * 16 + v + 8 * kk;
        float t = acc[v];
        // softplus = log1p(exp(t)); for t >> 0 exp overflows -> identity
        float sp = (t > 20.0f) ? t : log1pf(__expf(t));
        delta[(size_t)r * D_INNER + c] = sp;
    }
}

// ---------------------------------------------------------------------------
// Kernel 3: selective scan.
//   lanes 0-15  : 16 states of channel d0
//   lanes 16-31 : 16 states of channel d1
//   h[n] <- exp(delta*A[d,n]) * h[n] + delta * B[l,n] * x[l,d]
//   y[l,d] = sum_n h[n]*C[l,n] + D[d]*x[l,d]
// Each 256-thread block owns 16 consecutive channels -> delta/x loads for a
// step collapse to one 64B cacheline per block.  A row preloaded with log2(e)
// folded in so the per-step transcendental is a single v_exp_f32 (exp2f).
// ---------------------------------------------------------------------------
#define CH_PER_BLOCK 16

__global__ __launch_bounds__(256) void ssm_scan(
    const float* __restrict__ x,       // [L, D_INNER]
    const float* __restrict__ delta,   // [L, D_INNER]
    const float* __restrict__ Bm,      // [L, D_STATE]
    const float* __restrict__ Cm,      // [L, D_STATE]
    const float* __restrict__ A_log,   // [D_INNER, D_STATE]
    const float* __restrict__ Dv,      // [D_INNER]
    float* __restrict__ y)             // [L, D_INNER]
{
    const int lane = threadIdx.x & 31;
    const int n    = lane & 15;             // state index
    const int chw  = lane >> 4;             // which of 2 channels in this wave
    const int wave = threadIdx.x >> 5;      // 0..7
    const int d    = blockIdx.x * CH_PER_BLOCK + wave * 2 + chw;

    // A = -exp(A_log); fold log2(e) so the inner loop uses exp2f.
    const float Aln2 = -__expf(A_log[(size_t)d * D_STATE + n]) * 1.44269504088896340736f;
    const float Dd   = Dv[d];

    float h = 0.0f;
    for (int l = 0; l < SEQ_LEN; ++l) {
        const size_t ld = (size_t)l * D_INNER + d;
        const float dlt = delta[ld];
        const float xv  = x[ld];
        const float bv  = Bm[l * D_STATE + n];
        const float cv  = Cm[l * D_STATE + n];

        // prefetch the sequential streams a few lines ahead
        if (l + 16 < SEQ_LEN) {
            __builtin_prefetch(&delta[ld + (size_t)16 * D_INNER], 0, 0);
            __builtin_prefetch(&x[ld + (size_t)16 * D_INNER], 0, 0);
        }

        const float dA = exp2f(dlt * Aln2);          // v_exp_f32
        h = fmaf(dA, h, dlt * bv * xv);              // state update

        float p = h * cv;                            // dot(h, C) over 16 lanes
        p += __shfl_xor(p, 1, 32);
        p += __shfl_xor(p, 2, 32);
        p += __shfl_xor(p, 4, 32);
        p += __shfl_xor(p, 8, 32);

        if (n == 0) y[ld] = fmaf(Dd, xv, p);
    }
}

// ---------------------------------------------------------------------------
// Launch
// ---------------------------------------------------------------------------
extern "C" void kernel_launch(void* const* d_in, const int* in_sizes, int n_in,
                              void* d_out, int out_size, void* d_ws, size_t ws_size,
                              hipStream_t stream) {
    const float* x       = (const float*)d_in[0];  // [4096, 4096]
    const float* W_in    = (const float*)d_in[1];  // [160, 4096]
    const float* W_delta = (const float*)d_in[2];  // [4096, 128]
    const float* A_log   = (const float*)d_in[3];  // [4096, 16]
    const float* Dv      = (const float*)d_in[4];  // [4096]
    float* y             = (float*)d_out;          // [4096, 4096]

    // workspace layout
    float* dRaw  = (float*)d_ws;                                   // L*128
    float* Bm    = dRaw + (size_t)SEQ_LEN * DT_RANK;               // L*16
    float* Cm    = Bm   + (size_t)SEQ_LEN * D_STATE;               // L*16
    float* delta = Cm   + (size_t)SEQ_LEN * D_STATE;               // L*4096

    // Phase 1: input projection (WMMA fp32)
    ssm_gemm_in<<<dim3(SEQ_LEN / 16, NCOLS / 16), 32, 0, stream>>>(
        x, W_in, dRaw, Bm, Cm);

    // Phase 2: delta projection + softplus (WMMA fp32)
    ssm_gemm_delta<<<dim3(SEQ_LEN / 16, D_INNER / 16), 32, 0, stream>>>(
        dRaw, W_delta, delta);

    // Phase 3: fused selective scan
    ssm_scan<<<D_INNER / CH_PER_BLOCK, 256, 0, stream>>>(
        x, delta, Bm, Cm, A_log, Dv, y);
}